// GraphResBlock_18227841204838
// MI455X (gfx1250) — compile-verified
//
#include <hip/hip_runtime.h>
#include <math.h>

#define NN   65536
#define DD   128
#define EPSV 1e-5f

typedef __attribute__((ext_vector_type(2))) float v2f;
typedef __attribute__((ext_vector_type(8))) float v8f;

__device__ __forceinline__ float wred(float v) {
#pragma unroll
  for (int off = 16; off > 0; off >>= 1) v += __shfl_xor(v, off, 32);
  return v;
}

__device__ __forceinline__ float silu_f(float v) {
  return v / (1.0f + __expf(-v));
}

// ---------------------------------------------------------------------------
// K1: h1 = silu( layernorm(x)*n1w+n1b + cond @ cond_W + cond_b )
// one wave32 owns a 16-row tile; WMMA f32 16x16x4, 8 column tiles (full 128)
// ---------------------------------------------------------------------------
__global__ __launch_bounds__(128) void k1_ln_condgemm_silu(
    const float* __restrict__ x,     const float* __restrict__ cond,
    const float* __restrict__ condW, const float* __restrict__ condb,
    const float* __restrict__ n1w,   const float* __restrict__ n1b,
    float* __restrict__ h1)
{
  __shared__ float lnb[4][16][DD + 4];
  const int wv   = threadIdx.x >> 5;
  const int lane = threadIdx.x & 31;
  const int half = lane >> 4;        // 0: K pair {0,1}, 1: K pair {2,3}
  const int lrow = lane & 15;        // M (for A) / N (for B) within tile
  const int m0   = (blockIdx.x * 4 + wv) * 16;
  float (*tile)[DD + 4] = lnb[wv];

  // ---- layernorm(x) rows -> LDS --------------------------------------
  const int c4 = lane * 4;
  const float4 w1v = *reinterpret_cast<const float4*>(n1w + c4);
  const float4 b1v = *reinterpret_cast<const float4*>(n1b + c4);
#pragma unroll 4
  for (int r = 0; r < 16; ++r) {
    const float4 xv = *reinterpret_cast<const float4*>(x + (size_t)(m0 + r) * DD + c4);
    float s = xv.x + xv.y + xv.z + xv.w;
    float q = xv.x * xv.x + xv.y * xv.y + xv.z * xv.z + xv.w * xv.w;
    s = wred(s); q = wred(q);
    const float mu = s * (1.0f / DD);
    const float rs = rsqrtf(q * (1.0f / DD) - mu * mu + EPSV);
    tile[r][c4 + 0] = (xv.x - mu) * rs * w1v.x + b1v.x;
    tile[r][c4 + 1] = (xv.y - mu) * rs * w1v.y + b1v.y;
    tile[r][c4 + 2] = (xv.z - mu) * rs * w1v.z + b1v.z;
    tile[r][c4 + 3] = (xv.w - mu) * rs * w1v.w + b1v.w;
  }

  // ---- cond @ cond_W via WMMA f32 16x16x4 ----------------------------
  v8f acc[8];
  {
    v8f zero = {};
#pragma unroll
    for (int i = 0; i < 8; ++i) acc[i] = zero;
  }
  const float* arow = cond + (size_t)(m0 + lrow) * DD + half * 2;
  for (int kb = 0; kb < DD; kb += 4) {
    const float2 af = *reinterpret_cast<const float2*>(arow + kb);
    v2f a = {af.x, af.y};
    const int k0 = kb + half * 2;
    const float* bp0 = condW + (size_t)k0 * DD + lrow;
#pragma unroll
    for (int nt = 0; nt < 8; ++nt) {
      v2f b = {bp0[nt * 16], bp0[DD + nt * 16]};
      acc[nt] = __builtin_amdgcn_wmma_f32_16x16x4_f32(
          false, a, false, b, (short)0, acc[nt], false, false);
    }
  }

  // ---- epilogue: + ln + bias, SiLU, restage into LDS -----------------
  // each (row, col) element is owned by exactly one lane -> no barrier
#pragma unroll
  for (int nt = 0; nt < 8; ++nt) {
    const int c  = nt * 16 + lrow;
    const float cb = condb[c];
#pragma unroll
    for (int r = 0; r < 8; ++r) {
      const int rl = r + half * 8;
      tile[rl][c] = silu_f(acc[nt][r] + tile[rl][c] + cb);
    }
  }

  // ---- vectorized 128-bit stores of h1 -------------------------------
#pragma unroll 4
  for (int r = 0; r < 16; ++r) {
    float4 ov;
    ov.x = tile[r][c4 + 0]; ov.y = tile[r][c4 + 1];
    ov.z = tile[r][c4 + 2]; ov.w = tile[r][c4 + 3];
    *reinterpret_cast<float4*>(h1 + (size_t)(m0 + r) * DD + c4) = ov;
  }
}

// ---------------------------------------------------------------------------
// K2: z[row] += val * h1[col]  — 32 edges per wave32: coalesced index loads,
// shfl-broadcast per edge, float4/lane gather, 4 f32 atomic adds per lane
// ---------------------------------------------------------------------------
__global__ __launch_bounds__(256) void k2_spmm_scatter(
    const int* __restrict__ erow, const int* __restrict__ ecol,
    const float* __restrict__ ev, const float* __restrict__ h1,
    float* __restrict__ z, int E)
{
  const int lane = threadIdx.x & 31;
  const int wv   = threadIdx.x >> 5;
  const int e0   = (blockIdx.x * 8 + wv) * 32;
  if (e0 >= E) return;

  int   myr = 0, myc = 0;
  float myv = 0.0f;
  if (e0 + lane < E) {
    myr = erow[e0 + lane];
    myc = ecol[e0 + lane];
    myv = ev[e0 + lane];
  }

#pragma unroll 4
  for (int j = 0; j < 32; ++j) {
    if (e0 + j >= E) break;
    const int   r = __shfl(myr, j, 32);
    const int   c = __shfl(myc, j, 32);
    const float v = __shfl(myv, j, 32);
    const float4 hv = *reinterpret_cast<const float4*>(h1 + (size_t)c * DD + lane * 4);
    float* zp = z + (size_t)r * DD + lane * 4;
    atomicAdd(zp + 0, v * hv.x);
    atomicAdd(zp + 1, v * hv.y);
    atomicAdd(zp + 2, v * hv.z);
    atomicAdd(zp + 3, v * hv.w);
  }
}

// ---------------------------------------------------------------------------
// K3: out = x + (layernorm(z @ conv_W + conv_b)*n2w+n2b)*gamma + beta
// ---------------------------------------------------------------------------
__global__ __launch_bounds__(128) void k3_convgemm_ln_res(
    const float* __restrict__ z,     const float* __restrict__ convW,
    const float* __restrict__ convb, const float* __restrict__ n2w,
    const float* __restrict__ n2b,   const float* __restrict__ gam,
    const float* __restrict__ bet,   const float* __restrict__ x,
    float* __restrict__ out)
{
  __shared__ float gbuf[4][16][DD + 4];
  const int wv   = threadIdx.x >> 5;
  const int lane = threadIdx.x & 31;
  const int half = lane >> 4;
  const int lrow = lane & 15;
  const int m0   = (blockIdx.x * 4 + wv) * 16;
  float (*tile)[DD + 4] = gbuf[wv];

  // ---- z @ conv_W via WMMA f32 16x16x4 -------------------------------
  v8f acc[8];
  {
    v8f zero = {};
#pragma unroll
    for (int i = 0; i < 8; ++i) acc[i] = zero;
  }
  const float* arow = z + (size_t)(m0 + lrow) * DD + half * 2;
  for (int kb = 0; kb < DD; kb += 4) {
    const float2 af = *reinterpret_cast<const float2*>(arow + kb);
    v2f a = {af.x, af.y};
    const int k0 = kb + half * 2;
    const float* bp0 = convW + (size_t)k0 * DD + lrow;
#pragma unroll
    for (int nt = 0; nt < 8; ++nt) {
      v2f b = {bp0[nt * 16], bp0[DD + nt * 16]};
      acc[nt] = __builtin_amdgcn_wmma_f32_16x16x4_f32(
          false, a, false, b, (short)0, acc[nt], false, false);
    }
  }

  // ---- restage accumulators (+bias) row-major through LDS ------------
#pragma unroll
  for (int nt = 0; nt < 8; ++nt) {
    const int c  = nt * 16 + lrow;
    const float cb = convb[c];
#pragma unroll
    for (int r = 0; r < 8; ++r) tile[r + half * 8][c] = acc[nt][r] + cb;
  }

  // ---- per-row layernorm + affine + residual -------------------------
  const int c4 = lane * 4;
  const float4 w2v = *reinterpret_cast<const float4*>(n2w + c4);
  const float4 b2v = *reinterpret_cast<const float4*>(n2b + c4);
  const float4 gv  = *reinterpret_cast<const float4*>(gam + c4);
  const float4 bv  = *reinterpret_cast<const float4*>(bet + c4);
#pragma unroll 4
  for (int r = 0; r < 16; ++r) {
    float4 hv;
    hv.x = tile[r][c4 + 0]; hv.y = tile[r][c4 + 1];
    hv.z = tile[r][c4 + 2]; hv.w = tile[r][c4 + 3];
    float s = hv.x + hv.y + hv.z + hv.w;
    float q = hv.x * hv.x + hv.y * hv.y + hv.z * hv.z + hv.w * hv.w;
    s = wred(s); q = wred(q);
    const float mu = s * (1.0f / DD);
    const float rs = rsqrtf(q * (1.0f / DD) - mu * mu + EPSV);
    const float4 xv = *reinterpret_cast<const float4*>(x + (size_t)(m0 + r) * DD + c4);
    float4 ov;
    ov.x = xv.x + ((hv.x - mu) * rs * w2v.x + b2v.x) * gv.x + bv.x;
    ov.y = xv.y + ((hv.y - mu) * rs * w2v.y + b2v.y) * gv.y + bv.y;
    ov.z = xv.z + ((hv.z - mu) * rs * w2v.z + b2v.z) * gv.z + bv.z;
    ov.w = xv.w + ((hv.w - mu) * rs * w2v.w + b2v.w) * gv.w + bv.w;
    *reinterpret_cast<float4*>(out + (size_t)(m0 + r) * DD + c4) = ov;
  }
}

// ---------------------------------------------------------------------------
extern "C" void kernel_launch(void* const* d_in, const int* in_sizes, int n_in,
                              void* d_out, int out_size, void* d_ws, size_t ws_size,
                              hipStream_t stream) {
  const float* x     = (const float*)d_in[0];
  const int*   erow  = (const int*)d_in[1];
  const int*   ecol  = (const int*)d_in[2];
  const float* ev    = (const float*)d_in[3];
  const float* cond  = (const float*)d_in[4];
  const float* gam   = (const float*)d_in[5];
  const float* bet   = (const float*)d_in[6];
  const float* n1w   = (const float*)d_in[7];
  const float* n1b   = (const float*)d_in[8];
  const float* n2w   = (const float*)d_in[9];
  const float* n2b   = (const float*)d_in[10];
  const float* convW = (const float*)d_in[11];
  const float* convb = (const float*)d_in[12];
  const float* condW = (const float*)d_in[13];
  const float* condb = (const float*)d_in[14];
  float* out = (float*)d_out;

  float* h1 = (float*)d_ws;                       // [NN, DD]
  float* z  = h1 + (size_t)NN * DD;               // [NN, DD]
  const int E = in_sizes[1];

  hipMemsetAsync(z, 0, (size_t)NN * DD * sizeof(float), stream);

  k1_ln_condgemm_silu<<<NN / 64, 128, 0, stream>>>(x, cond, condW, condb, n1w, n1b, h1);
  k2_spmm_scatter<<<(E + 255) / 256, 256, 0, stream>>>(erow, ecol, ev, h1, z, E);
  k3_convgemm_ln_res<<<NN / 64, 128, 0, stream>>>(z, convW, convb, n2w, n2b, gam, bet, x, out);
}